// StackedSTU2D_48550310314383
// MI455X (gfx1250) — compile-verified
//
#include <hip/hip_runtime.h>
#include <math.h>

// ---------------- problem constants ----------------
#define NB 4
#define ND 64
#define NX 128
#define NY 128
#define NL 8
#define NDEPTH 4
#define YRR 65            // rfft length along y
#define YRP 80            // padded (multiple of 16) frequency dim
#define NPIX (NX*NY)      // 16384
#define FDIM (NX*YRP)     // 10240 flattened padded frequencies
#define SLABS (NB*ND)     // 256
#define ROWS (NB*ND*NX)   // 32768
#define PI_D 3.14159265358979323846

typedef float v2f __attribute__((ext_vector_type(2)));
typedef float v8f __attribute__((ext_vector_type(8)));

#define V8Z {0.f,0.f,0.f,0.f,0.f,0.f,0.f,0.f}

// D = A(16x4) * B(4x16) + C, f32 WMMA
#define WMMA4(acc, a, b) \
  (acc) = __builtin_amdgcn_wmma_f32_16x16x4_f32(false, (a), false, (b), (short)0, (acc), false, false)

// A fragment: lane -> m = lane&15 ; k = 2*(lane>>4) + v  (contiguous pair -> b64 load)
static __device__ __forceinline__ v2f afrag(const float* __restrict__ p, int lda, int lane) {
  const float* q = p + (lane & 15) * lda + 2 * (lane >> 4);
  v2f r; r.x = q[0]; r.y = q[1];
  return r;
}
// B fragment: lane -> n = lane&15 ; k = 2*(lane>>4) + v
static __device__ __forceinline__ v2f bfrag(const float* __restrict__ p, int ldb, int lane) {
  const float* q = p + (lane & 15) + (2 * (lane >> 4)) * ldb;
  v2f r; r.x = q[0]; r.y = q[ldb];
  return r;
}
// C/D fragment store: lane -> n = lane&15 ; row = v + 8*(lane>>4)
static __device__ __forceinline__ void cstore(float* __restrict__ p, int ldd, int lane, v8f c) {
  float* q = p + (lane & 15) + (8 * (lane >> 4)) * ldd;
#pragma unroll
  for (int v = 0; v < 8; ++v) q[v * ldd] = c[v];
}

// ---------------- DFT matrix tables ----------------
// Cy [y=128][k=80]   : cos(2*pi*y*k/128)                (forward rfft_y, real part)
// Syn[y=128][k=80]   : -sin(...)                        (forward rfft_y, imag part)
// Cx [128][128]      : cos(2*pi*r*c/128)  (symmetric, shared fwd/inv along x)
// Sxp/Sxn [128][128] : +sin / -sin
// ICy[k=80][y=128]   : w_k*cos(2*pi*k*y/128), w = {1,2,..,2,1}/16384 (full irfftn norm)
// ISn[k=80][y=128]   : -w_k*sin(...)
__global__ void k_init(float* Cy, float* Syn, float* Cx, float* Sxp, float* Sxn,
                       float* ICy, float* ISn) {
  int idx = blockIdx.x * blockDim.x + threadIdx.x;
  if (idx < NX * NX) {
    int r = idx >> 7, c = idx & 127;
    double a = (2.0 * PI_D / 128.0) * (double)((r * c) & 127);
    float cv = (float)cos(a), sv = (float)sin(a);
    Cx[idx] = cv; Sxp[idx] = sv; Sxn[idx] = -sv;
  }
  if (idx < NX * YRP) {
    int y = idx / YRP, k = idx % YRP;
    if (k < YRR) {
      double a = (2.0 * PI_D / 128.0) * (double)((y * k) & 127);
      Cy[idx] = (float)cos(a);
      Syn[idx] = -(float)sin(a);
    } else { Cy[idx] = 0.f; Syn[idx] = 0.f; }
    int k2 = idx >> 7, y2 = idx & 127;
    if (k2 < YRR) {
      double w = ((k2 == 0) || (k2 == 64)) ? (1.0 / 16384.0) : (2.0 / 16384.0);
      double a2 = (2.0 * PI_D / 128.0) * (double)((k2 * y2) & 127);
      ICy[idx] = (float)(w * cos(a2));
      ISn[idx] = (float)(-w * sin(a2));
    } else { ICy[idx] = 0.f; ISn[idx] = 0.f; }
  }
}

// ---------------- lift: concat grid channels + pointwise conv ----------------
__global__ void k_lift(const float* __restrict__ X, const float* __restrict__ lw,
                       const float* __restrict__ lb, float* __restrict__ H) {
  int idx = blockIdx.x * blockDim.x + threadIdx.x;
  if (idx >= NB * ND * NPIX) return;
  int pix = idx & (NPIX - 1);
  int d = (idx >> 14) & (ND - 1);
  int b = idx >> 20;
  int xr = pix >> 7, yc = pix & 127;
  const float* xb = X + (size_t)b * 3 * NPIX + pix;
  const float* w = lw + d * 5;
  float acc = lb[d];
  acc += w[0] * xb[0] + w[1] * xb[NPIX] + w[2] * xb[2 * NPIX];
  acc += w[3] * ((float)xr * (1.0f / 127.0f));   // yy channel = ys[x]
  acc += w[4] * ((float)yc * (1.0f / 127.0f));   // xx channel = xs[y]
  H[idx] = acc;
}

// ---------------- LayerNorm over channel dim ----------------
__global__ void k_ln(const float* __restrict__ H, const float* __restrict__ g,
                     const float* __restrict__ bta, float* __restrict__ Z) {
  int idx = blockIdx.x * blockDim.x + threadIdx.x;
  if (idx >= NB * NPIX) return;
  int b = idx >> 14, pix = idx & (NPIX - 1);
  const float* hp = H + (size_t)b * ND * NPIX + pix;
  float s = 0.f, ss = 0.f;
#pragma unroll 8
  for (int d = 0; d < ND; ++d) { float v = hp[d * NPIX]; s += v; ss += v * v; }
  float mu = s * (1.f / (float)ND);
  float var = ss * (1.f / (float)ND) - mu * mu;
  float rs = rsqrtf(var + 1e-5f);
  float* zp = Z + (size_t)b * ND * NPIX + pix;
#pragma unroll 8
  for (int d = 0; d < ND; ++d) {
    float v = hp[d * NPIX];
    zp[d * NPIX] = (v - mu) * rs * g[d] + bta[d];
  }
}

// ---------------- forward rfft along y as GEMM ----------------
// rows = (b,d,x) flattened (32768), K = y (128), N = padded freq (80)
// 4 M-tiles per wave: DFT B-fragments loaded once per k-step, reused 4x
__global__ __launch_bounds__(32) void k_yfwd(const float* __restrict__ Z,
    const float* __restrict__ Cy, const float* __restrict__ Syn,
    float* __restrict__ Yr, float* __restrict__ Yi) {
  const int lane = threadIdx.x;
  const int nt = blockIdx.x;
  const int m0 = blockIdx.y * 64;          // 4 row tiles
  const float* A0 = Z + (size_t)m0 * NY;
  const float* Bc = Cy + nt * 16;
  const float* Bs = Syn + nt * 16;
  v8f ar[4] = {V8Z, V8Z, V8Z, V8Z};
  v8f ai[4] = {V8Z, V8Z, V8Z, V8Z};
#pragma unroll 2
  for (int k0 = 0; k0 < NY; k0 += 4) {
    v2f br = bfrag(Bc + k0 * YRP, YRP, lane);
    v2f bi = bfrag(Bs + k0 * YRP, YRP, lane);
#pragma unroll
    for (int j = 0; j < 4; ++j) {
      v2f a = afrag(A0 + (size_t)j * 16 * NY + k0, NY, lane);
      WMMA4(ar[j], a, br);
      WMMA4(ai[j], a, bi);
    }
  }
#pragma unroll
  for (int j = 0; j < 4; ++j) {
    cstore(Yr + (size_t)(m0 + j * 16) * YRP + nt * 16, YRP, lane, ar[j]);
    cstore(Yi + (size_t)(m0 + j * 16) * YRP + nt * 16, YRP, lane, ai[j]);
  }
}

// ---------------- complex DFT along x as GEMM (forward & inverse) ----------------
// Or = C*Ir + SA*Ii ; Oi = SB*Ir + C*Ii
// forward: SA=+sin, SB=-sin ; inverse: SA=-sin, SB=+sin (1/N folded into ICy)
// 4 M-tiles per wave: strided data B-fragments loaded once, reused across 4 DFT row-tiles
__global__ __launch_bounds__(32) void k_xform(const float* __restrict__ Ir,
    const float* __restrict__ Ii, const float* __restrict__ C,
    const float* __restrict__ SA, const float* __restrict__ SB,
    float* __restrict__ Or_, float* __restrict__ Oi_) {
  const int lane = threadIdx.x;
  const int nt = blockIdx.x, slab = blockIdx.z;
  const int m0 = blockIdx.y * 64;          // 4 row tiles of the 128 kx rows
  const float* br0 = Ir + (size_t)slab * FDIM + nt * 16;
  const float* bi0 = Ii + (size_t)slab * FDIM + nt * 16;
  v8f accR[4] = {V8Z, V8Z, V8Z, V8Z};
  v8f accI[4] = {V8Z, V8Z, V8Z, V8Z};
#pragma unroll 2
  for (int k0 = 0; k0 < NX; k0 += 4) {
    v2f br = bfrag(br0 + k0 * YRP, YRP, lane);
    v2f bi = bfrag(bi0 + k0 * YRP, YRP, lane);
#pragma unroll
    for (int j = 0; j < 4; ++j) {
      const int ro = (m0 + j * 16) * NX + k0;
      v2f ac = afrag(C + ro, NX, lane);
      v2f aa = afrag(SA + ro, NX, lane);
      v2f ab = afrag(SB + ro, NX, lane);
      WMMA4(accR[j], ac, br);
      WMMA4(accR[j], aa, bi);
      WMMA4(accI[j], ab, br);
      WMMA4(accI[j], ac, bi);
    }
  }
#pragma unroll
  for (int j = 0; j < 4; ++j) {
    cstore(Or_ + (size_t)slab * FDIM + (size_t)(m0 + j * 16) * YRP + nt * 16, YRP, lane, accR[j]);
    cstore(Oi_ + (size_t)slab * FDIM + (size_t)(m0 + j * 16) * YRP + nt * 16, YRP, lane, accI[j]);
  }
}

// ---------------- spectral contraction ----------------
// Sf[b,h,f] = sum_l conj(Phi[l,f]) * ( sum_d Theta[l,h,d] * Zf[b,d,f] )
// 4 N-tiles (frequency) per wave: Theta A-fragment loaded once per k-step, feeds 8 WMMAs
__global__ __launch_bounds__(32) void k_contract(const float* __restrict__ Zr,
    const float* __restrict__ Zi, const float* __restrict__ Th,
    const float* __restrict__ Phi, float* __restrict__ Sr, float* __restrict__ Si) {
  const int lane = threadIdx.x;
  const int ft4 = blockIdx.x, mt = blockIdx.y, b = blockIdx.z;
  const int f0 = ft4 * 64;
  const float* br0 = Zr + (size_t)b * (ND * FDIM) + f0;
  const float* bi0 = Zi + (size_t)b * (ND * FDIM) + f0;
  v8f sr[4] = {V8Z, V8Z, V8Z, V8Z};
  v8f si[4] = {V8Z, V8Z, V8Z, V8Z};
  for (int l = 0; l < NL; ++l) {
    const float* A0 = Th + (size_t)(l * ND + mt * 16) * ND;
    v8f tr[4] = {V8Z, V8Z, V8Z, V8Z};
    v8f ti[4] = {V8Z, V8Z, V8Z, V8Z};
#pragma unroll 2
    for (int k0 = 0; k0 < ND; k0 += 4) {
      v2f a = afrag(A0 + k0, ND, lane);
#pragma unroll
      for (int j = 0; j < 4; ++j) {
        v2f br = bfrag(br0 + (size_t)k0 * FDIM + j * 16, FDIM, lane);
        v2f bi = bfrag(bi0 + (size_t)k0 * FDIM + j * 16, FDIM, lane);
        WMMA4(tr[j], a, br);
        WMMA4(ti[j], a, bi);
      }
    }
#pragma unroll
    for (int j = 0; j < 4; ++j) {
      const int f = f0 + j * 16 + (lane & 15);
      const int kx = f / YRP, kf = f % YRP;
      float pr = 0.f, pi = 0.f;
      if (kf < YRR) {  // padded columns get zero filter -> stay exactly zero
        const float* pp = Phi + (size_t)((l * NX + kx) * YRR + kf) * 2;
        pr = pp[0]; pi = pp[1];
      }
      // conj(P)*T : Sr += pr*Tr + pi*Ti ; Si += pr*Ti - pi*Tr (pr/pi uniform per lane column)
#pragma unroll
      for (int v = 0; v < 8; ++v) {
        sr[j][v] += pr * tr[j][v] + pi * ti[j][v];
        si[j][v] += pr * ti[j][v] - pi * tr[j][v];
      }
    }
  }
#pragma unroll
  for (int j = 0; j < 4; ++j) {
    cstore(Sr + (size_t)b * (ND * FDIM) + (size_t)mt * 16 * FDIM + f0 + j * 16, FDIM, lane, sr[j]);
    cstore(Si + (size_t)b * (ND * FDIM) + (size_t)mt * 16 * FDIM + f0 + j * 16, FDIM, lane, si[j]);
  }
}

// ---------------- inverse rfft along y as GEMM (real output) ----------------
// rows = (b,h,x) (32768), K = padded freq (80, zero-weighted tail), N = y (128)
// 4 M-tiles per wave: ICy/ISn B-fragments reused 4x
__global__ __launch_bounds__(32) void k_yinv(const float* __restrict__ S1r,
    const float* __restrict__ S1i, const float* __restrict__ ICy,
    const float* __restrict__ ISn, float* __restrict__ Out) {
  const int lane = threadIdx.x;
  const int nt = blockIdx.x;
  const int m0 = blockIdx.y * 64;
  const float* b10 = ICy + nt * 16;
  const float* b20 = ISn + nt * 16;
  v8f acc[4] = {V8Z, V8Z, V8Z, V8Z};
#pragma unroll 2
  for (int k0 = 0; k0 < YRP; k0 += 4) {
    v2f b1 = bfrag(b10 + k0 * NY, NY, lane);
    v2f b2 = bfrag(b20 + k0 * NY, NY, lane);
#pragma unroll
    for (int j = 0; j < 4; ++j) {
      v2f a1 = afrag(S1r + (size_t)(m0 + j * 16) * YRP + k0, YRP, lane);
      v2f a2 = afrag(S1i + (size_t)(m0 + j * 16) * YRP + k0, YRP, lane);
      WMMA4(acc[j], a1, b1);
      WMMA4(acc[j], a2, b2);
    }
  }
#pragma unroll
  for (int j = 0; j < 4; ++j)
    cstore(Out + (size_t)(m0 + j * 16) * NY + nt * 16, NY, lane, acc[j]);
}

// ---------------- GLU + residual (two fused GEMMs + sigmoid) ----------------
// 4 pixel-tiles per wave: vw/gw A-fragments reused 4x
__global__ __launch_bounds__(32) void k_glu(const float* __restrict__ S,
    const float* __restrict__ vw, const float* __restrict__ vb,
    const float* __restrict__ gw, const float* __restrict__ gb,
    float* __restrict__ H) {
  const int lane = threadIdx.x;
  const int nt4 = blockIdx.x, mt = blockIdx.y, b = blockIdx.z;
  const int n0 = nt4 * 64;
  const float* B0 = S + (size_t)b * (ND * NPIX) + n0;
  const float* Av0 = vw + mt * 16 * ND;
  const float* Ag0 = gw + mt * 16 * ND;
  v8f av[4] = {V8Z, V8Z, V8Z, V8Z};
  v8f ag[4] = {V8Z, V8Z, V8Z, V8Z};
#pragma unroll 2
  for (int k0 = 0; k0 < ND; k0 += 4) {
    v2f a1 = afrag(Av0 + k0, ND, lane);
    v2f a2 = afrag(Ag0 + k0, ND, lane);
#pragma unroll
    for (int j = 0; j < 4; ++j) {
      v2f bf = bfrag(B0 + (size_t)k0 * NPIX + j * 16, NPIX, lane);
      WMMA4(av[j], a1, bf);
      WMMA4(ag[j], a2, bf);
    }
  }
  const int n = lane & 15, g = lane >> 4;
#pragma unroll
  for (int j = 0; j < 4; ++j) {
    float* Hp = H + (size_t)b * (ND * NPIX) + n0 + j * 16 + n;
#pragma unroll
    for (int v = 0; v < 8; ++v) {
      int o = mt * 16 + v + 8 * g;
      float val = av[j][v] + vb[o];
      float gate = ag[j][v] + gb[o];
      gate = 1.0f / (1.0f + __expf(-gate));
      Hp[(size_t)o * NPIX] += val * gate;
    }
  }
}

// ---------------- head: OUT=1 dot per pixel ----------------
__global__ void k_head(const float* __restrict__ H, const float* __restrict__ hw,
                       const float* __restrict__ hb, float* __restrict__ Out) {
  int idx = blockIdx.x * blockDim.x + threadIdx.x;
  if (idx >= NB * NPIX) return;
  int b = idx >> 14, pix = idx & (NPIX - 1);
  const float* hp = H + (size_t)b * ND * NPIX + pix;
  float acc = hb[0];
#pragma unroll 8
  for (int d = 0; d < ND; ++d) acc += hw[d] * hp[d * NPIX];
  Out[idx] = acc;
}

// ---------------- launch ----------------
extern "C" void kernel_launch(void* const* d_in, const int* in_sizes, int n_in,
                              void* d_out, int out_size, void* d_ws, size_t ws_size,
                              hipStream_t stream) {
  (void)in_sizes; (void)n_in; (void)out_size; (void)ws_size;
  const float* x      = (const float*)d_in[0];
  const float* Phi    = (const float*)d_in[1];   // complex64 interleaved (L,128,65)
  const float* lift_w = (const float*)d_in[2];
  const float* lift_b = (const float*)d_in[3];
  const float* Theta  = (const float*)d_in[4];   // (DEPTH,L,D,D)
  const float* vW     = (const float*)d_in[5];
  const float* vB     = (const float*)d_in[6];
  const float* gW     = (const float*)d_in[7];
  const float* gB     = (const float*)d_in[8];
  const float* lnG    = (const float*)d_in[9];
  const float* lnB    = (const float*)d_in[10];
  const float* hW     = (const float*)d_in[11];
  const float* hB     = (const float*)d_in[12];
  float* out = (float*)d_out;

  float* W = (float*)d_ws;
  size_t o = 0;
  float* H   = W + o; o += (size_t)NB * ND * NPIX;   // activations (residual stream)
  float* Zb  = W + o; o += (size_t)NB * ND * NPIX;   // z / spatial S (ping)
  float* Ar  = W + o; o += (size_t)SLABS * FDIM;     // Yr / Sr
  float* Ai  = W + o; o += (size_t)SLABS * FDIM;     // Yi / Si
  float* Br  = W + o; o += (size_t)SLABS * FDIM;     // Zr / S1r
  float* Bi  = W + o; o += (size_t)SLABS * FDIM;     // Zi / S1i
  float* Cy  = W + o; o += NX * YRP;
  float* Syn = W + o; o += NX * YRP;
  float* Cx  = W + o; o += NX * NX;
  float* Sxp = W + o; o += NX * NX;
  float* Sxn = W + o; o += NX * NX;
  float* ICy = W + o; o += YRP * NY;
  float* ISn = W + o; o += YRP * NY;

  k_init<<<dim3((NX * NX + 255) / 256), dim3(256), 0, stream>>>(Cy, Syn, Cx, Sxp, Sxn, ICy, ISn);
  k_lift<<<dim3((NB * ND * NPIX + 255) / 256), dim3(256), 0, stream>>>(x, lift_w, lift_b, H);

  for (int layer = 0; layer < NDEPTH; ++layer) {
    k_ln<<<dim3(NB * NPIX / 256), dim3(256), 0, stream>>>(
        H, lnG + layer * ND, lnB + layer * ND, Zb);
    k_yfwd<<<dim3(YRP / 16, ROWS / 64), dim3(32), 0, stream>>>(Zb, Cy, Syn, Ar, Ai);
    k_xform<<<dim3(YRP / 16, NX / 64, SLABS), dim3(32), 0, stream>>>(
        Ar, Ai, Cx, Sxp, Sxn, Br, Bi);                       // forward fft_x
    k_contract<<<dim3(FDIM / 64, ND / 16, NB), dim3(32), 0, stream>>>(
        Br, Bi, Theta + (size_t)layer * NL * ND * ND, Phi, Ar, Ai);
    k_xform<<<dim3(YRP / 16, NX / 64, SLABS), dim3(32), 0, stream>>>(
        Ar, Ai, Cx, Sxn, Sxp, Br, Bi);                       // inverse ifft_x
    k_yinv<<<dim3(NY / 16, ROWS / 64), dim3(32), 0, stream>>>(Br, Bi, ICy, ISn, Zb);
    k_glu<<<dim3(NPIX / 64, ND / 16, NB), dim3(32), 0, stream>>>(
        Zb, vW + (size_t)layer * ND * ND, vB + layer * ND,
        gW + (size_t)layer * ND * ND, gB + layer * ND, H);
  }
  k_head<<<dim3(NB * NPIX / 256), dim3(256), 0, stream>>>(H, hW, hB, out);
}